// TMM_predictor_22651657519179
// MI455X (gfx1250) — compile-verified
//
#include <hip/hip_runtime.h>
#include <cstdint>

// TMM predictor for MI455X (gfx1250, wave32).
// Compute-bound complex 2x2 recurrence -> no matmul structure (WMMA N/A);
// uses CDNA5 async global->LDS pipeline (ASYNCcnt) for operand staging.

#define NUM_LAYERS 18
#define NIN        16      // inner layers (recurrence length)
#define LAM_TILE   32      // wavelengths per block
#define F_TILE     8       // d-rows per block
#define THREADS    256     // 8 waves (wave32)
// Record stride 70 floats: lane l -> bank 6l mod 64 (injective over 32 lanes),
// low/high dwords of b64 reads cover even/odd banks -> conflict-free.
#define REC_STR    70

typedef int v4i_t __attribute__((vector_size(16)));
typedef __attribute__((address_space(1))) v4i_t* g128_p;   // global 16B chunk
typedef __attribute__((address_space(3))) v4i_t* l128_p;   // LDS    16B chunk

__device__ __forceinline__ void async_ld_b128(const void* g, void* l) {
#if __has_builtin(__builtin_amdgcn_global_load_async_to_lds_b128)
  __builtin_amdgcn_global_load_async_to_lds_b128(
      (g128_p)(uintptr_t)g, (l128_p)l, /*ioffset=*/0, /*cpol=*/0);
#else
  unsigned lds_off = (unsigned)(uintptr_t)(l128_p)l;
  asm volatile("global_load_async_to_lds_b128 %0, %1, off"
               : : "v"(lds_off), "v"(g) : "memory");
#endif
}

__device__ __forceinline__ void wait_async0() {
#if __has_builtin(__builtin_amdgcn_s_wait_asynccnt)
  __builtin_amdgcn_s_wait_asynccnt(0);
#else
  asm volatile("s_wait_asynccnt 0" : : : "memory");
#endif
}

__global__ __launch_bounds__(THREADS)
void tmm_kernel(const float* __restrict__ d_arr,    // (num_f, 16)
                const float* __restrict__ lam_vac,  // (num_lam)
                const float* __restrict__ n_real,   // (num_lam, 18)
                const float* __restrict__ n_imag,   // (num_lam, 18)
                float* __restrict__ out,            // (num_f, num_lam)
                int num_f, int num_lam)
{
  // LDS layout (floats):
  //  [0,32)      lam_vac chunk
  //  [32,608)    n_real chunk  (32 x 18)
  //  [608,1184)  n_imag chunk  (32 x 18)
  //  [1184,1312) d rows        (8 x 16)
  //  [1312,3552) derived per-lam records (32 x 70), 8B-aligned records
  __shared__ float lds[3552];
  float* s_lam = lds;
  float* s_nre = lds + 32;
  float* s_nim = lds + 608;
  float* s_d   = lds + 1184;
  float* s_rec = lds + 1312;

  const int tid      = threadIdx.x;
  const int lam_base = blockIdx.x * LAM_TILE;
  const int f_base   = blockIdx.y * F_TILE;
  const int lam_cnt  = (lam_base + LAM_TILE <= num_lam) ? LAM_TILE : (num_lam - lam_base);

  // ---- Stage raw inputs into LDS via async B128 copies (ASYNCcnt path) ----
  {
    const int nch_lam = lam_cnt >> 2;                  // 4 floats / 16B chunk
    const int nch_n   = (lam_cnt * NUM_LAYERS) >> 2;   // 18*lam_cnt floats (lam_cnt % 4 == 0)
    const int nch_d   = (F_TILE * NIN) >> 2;           // 32 chunks

    const char* g_lam = (const char*)(lam_vac + lam_base);
    const char* g_nre = (const char*)(n_real + (size_t)lam_base * NUM_LAYERS);
    const char* g_nim = (const char*)(n_imag + (size_t)lam_base * NUM_LAYERS);
    const char* g_d   = (const char*)(d_arr  + (size_t)f_base * NIN);

    for (int i = tid; i < nch_lam; i += THREADS)
      async_ld_b128(g_lam + i * 16, (char*)s_lam + i * 16);
    for (int i = tid; i < nch_n; i += THREADS)
      async_ld_b128(g_nre + i * 16, (char*)s_nre + i * 16);
    for (int i = tid; i < nch_n; i += THREADS)
      async_ld_b128(g_nim + i * 16, (char*)s_nim + i * 16);
    for (int i = tid; i < nch_d; i += THREADS)
      async_ld_b128(g_d + i * 16, (char*)s_d + i * 16);

    wait_async0();        // each wave drains its own async loads
  }
  __syncthreads();        // make LDS visible across waves

  // ---- Derive per-lambda records (32 lanes, once per block) ----
  // Record (floats): [2j,2j+1]   = (kz_re, kz_im)  inner layer j   (j=0..15)
  //                  [32+2j,+1]  = (r_re, r_im)    interface j+1   (j=0..15)
  //                  [64,65]     = (r0_re, r0_im)
  //                  [66]        = S = |t0|^2 * prod|t_l|^2 * Re(nL/n0)
  if (tid < lam_cnt) {
    const float* nr  = s_nre + tid * NUM_LAYERS;
    const float* ni  = s_nim + tid * NUM_LAYERS;
    float*       rec = s_rec + tid * REC_STR;

    const float scale = 6.2831853071795864769f / s_lam[tid];  // 2*pi/lam
    #pragma unroll
    for (int j = 0; j < NIN; ++j) {
      *(float2*)(rec + 2 * j) = make_float2(scale * nr[j + 1], scale * ni[j + 1]);
    }

    float prodT = 1.0f, r0re = 0.0f, r0im = 0.0f;
    #pragma unroll
    for (int i = 0; i < NUM_LAYERS - 1; ++i) {
      const float lor = nr[i],     loi = ni[i];
      const float hir = nr[i + 1], hii = ni[i + 1];
      const float dre = lor + hir, dim_ = loi + hii;
      const float inv = 1.0f / (dre * dre + dim_ * dim_);
      const float numr = lor - hir, numi = loi - hii;
      const float rre = (numr * dre + numi * dim_) * inv;
      const float rim = (numi * dre - numr * dim_) * inv;
      const float t2  = 4.0f * (lor * lor + loi * loi) * inv;  // |t_i|^2
      if (i == 0) { r0re = rre; r0im = rim; prodT = t2; }
      else {
        *(float2*)(rec + 32 + 2 * (i - 1)) = make_float2(rre, rim);
        prodT *= t2;
      }
    }
    const float n0r = nr[0], n0i = ni[0];
    const float nLr = nr[NUM_LAYERS - 1], nLi = ni[NUM_LAYERS - 1];
    const float rr_re = (nLr * n0r + nLi * n0i) / (n0r * n0r + n0i * n0i);
    *(float2*)(rec + 64) = make_float2(r0re, r0im);
    rec[66] = prodT * rr_re;
  }
  __syncthreads();

  // ---- Main recurrence: one thread per (f, lambda) ----
  const int lam_sub = tid & 31;     // fast index -> coalesced stores
  const int f_sub   = tid >> 5;     // whole wave shares one f -> broadcast LDS reads
  const int f       = f_base + f_sub;
  if (lam_sub < lam_cnt && f < num_f) {
    const float* rec = s_rec + lam_sub * REC_STR;

    // Preload this thread's d row once: 4 broadcast ds_load_b128 -> registers.
    float dreg[NIN];
    {
      const float4* dv = (const float4*)(s_d + f_sub * NIN);
      #pragma unroll
      for (int k = 0; k < 4; ++k) {
        const float4 t = dv[k];
        dreg[4 * k + 0] = t.x; dreg[4 * k + 1] = t.y;
        dreg[4 * k + 2] = t.z; dreg[4 * k + 3] = t.w;
      }
    }

    // v = [1, r0];  v <- v * [[em, r*em],[r*ep, ep]]  (1/t factored into S)
    const float2 r0 = *(const float2*)(rec + 64);
    float pr = 1.0f, pi = 0.0f;
    float qr = r0.x, qi = r0.y;
    #pragma unroll
    for (int j = 0; j < NIN; ++j) {
      const float2 kz = *(const float2*)(rec + 2 * j);        // ds_load_b64
      const float2 rv = *(const float2*)(rec + 32 + 2 * j);   // ds_load_b64
      const float dj = dreg[j];
      const float ph = dj * kz.x;        // Re(delta)
      const float at = dj * kz.y;        // Im(delta)
      float sn, cs;
      __sincosf(ph, &sn, &cs);
      const float g  = __expf(at);                 // |exp(-i*delta)|
      const float gi = __builtin_amdgcn_rcpf(g);   // |exp(+i*delta)|
      // u = p*em, w = q*ep  with em=g*(cs - i*sn), ep=gi*(cs + i*sn)
      const float ur = g  * (pr * cs + pi * sn);
      const float ui = g  * (pi * cs - pr * sn);
      const float wr = gi * (qr * cs - qi * sn);
      const float wi = gi * (qi * cs + qr * sn);
      pr = ur + (rv.x * wr - rv.y * wi);
      pi = ui + (rv.x * wi + rv.y * wr);
      qr = wr + (rv.x * ur - rv.y * ui);
      qi = wi + (rv.x * ui + rv.y * ur);
    }
    const float denom = pr * pr + pi * pi;   // |a + r0*c|^2
    out[(size_t)f * num_lam + lam_base + lam_sub] = rec[66] / denom;
  }
}

extern "C" void kernel_launch(void* const* d_in, const int* in_sizes, int n_in,
                              void* d_out, int out_size, void* d_ws, size_t ws_size,
                              hipStream_t stream) {
  const float* d_arr   = (const float*)d_in[0];  // (num_f, 16)
  const float* lam_vac = (const float*)d_in[1];  // (num_lam)
  const float* n_real  = (const float*)d_in[2];  // (num_lam, 18)
  const float* n_imag  = (const float*)d_in[3];  // (num_lam, 18)
  float* out = (float*)d_out;

  const int num_f   = in_sizes[0] / NIN;   // 2048
  const int num_lam = in_sizes[1];         // 1000

  dim3 grid((num_lam + LAM_TILE - 1) / LAM_TILE,   // 32
            (num_f + F_TILE - 1) / F_TILE);        // 256
  tmm_kernel<<<grid, THREADS, 0, stream>>>(d_arr, lam_vac, n_real, n_imag,
                                           out, num_f, num_lam);
}